// EdgeGatedSAGELayer_15006615732398
// MI455X (gfx1250) — compile-verified
//
#include <hip/hip_runtime.h>
#include <cstdint>

// ---------------------------------------------------------------------------
// EdgeGatedSAGE layer for MI455X (gfx1250, wave32, WGP w/ 320KB LDS).
//
// All GEMMs use V_WMMA_F32_16X16X4_F32 (full f32 precision; workload is
// L2/atomic bound at ~0.5 GB effective HBM traffic vs ~53 GF, so f32 matrix
// throughput is ample). Weights are staged *transposed* in LDS once per
// persistent block (padded stride -> conflict-free ds_load_b64 B-fragments),
// A tiles are held in registers across the N loop, so:
//   - W traffic:  grid_blocks * 64KB   (was ~4 GB of redundant global reloads)
//   - A traffic:  each gathered row read exactly once (L2-resident, 51MB)
//   - scatter:    f32 atomics resolve in the 192MB L2
// ---------------------------------------------------------------------------

typedef float v2f __attribute__((ext_vector_type(2)));
typedef float v8f __attribute__((ext_vector_type(8)));

#define D_MODEL   128
#define EDGE_DIM  64
#define LN_EPS    1e-5f
#define LDS_STR_G 66    // 64 + 2 pad: bank = 2n mod 64, conflict-free b64
#define LDS_STR_W 130   // 128 + 2 pad

static __device__ __forceinline__ v8f wmma_f32_16x16x4(v2f a, v2f b, v8f c) {
  // 8 args: (neg_a, A, neg_b, B, c_mod, C, reuse_a, reuse_b)
  return __builtin_amdgcn_wmma_f32_16x16x4_f32(false, a, false, b, (short)0, c,
                                               false, false);
}

static __device__ __forceinline__ float gelu_exact(float x) {
  return 0.5f * x * (1.0f + erff(x * 0.70710678118654752f));
}

static __device__ __forceinline__ float sigmoidf(float x) {
  return 1.0f / (1.0f + __expf(-x));
}

// ---------------------------------------------------------------------------
__global__ void k_zero(float* __restrict__ p, long long n) {
  long long i = (long long)blockIdx.x * blockDim.x + threadIdx.x;
  if (i < n) p[i] = 0.0f;
}

// ---------------------------------------------------------------------------
// gate[e] = sigmoid( gelu(edge_attr[e] @ Wg1 + bg1) @ Wg2 + bg2 )
// Persistent waves, one 16-edge tile per wave iteration. Wg1 (64x128) staged
// transposed in LDS; A tile (16x64) held in registers across N-tiles.
__global__ void k_edge_gate(const float* __restrict__ edge_attr,
                            const float* __restrict__ Wg1,
                            const float* __restrict__ bg1,
                            const float* __restrict__ Wg2,
                            const float* __restrict__ bg2,
                            float* __restrict__ gate, int n_edges) {
  __shared__ float Wt[D_MODEL * LDS_STR_G];  // Wt[n*66 + k] = Wg1[k][n]
  const int tid = threadIdx.x;
  for (int i = tid; i < EDGE_DIM * D_MODEL; i += blockDim.x) {
    const int k = i >> 7, n = i & 127;
    Wt[n * LDS_STR_G + k] = Wg1[i];
  }
  __syncthreads();

  const int lane = tid & 31;
  const int wave = tid >> 5;
  const int wpb = blockDim.x >> 5;
  const int tiles = (n_edges + 15) >> 4;
  const int m = lane & 15;
  const int half = lane >> 4;
  const int koff = half ? 2 : 0;  // A/B K-row split across lane halves
  const float b2 = bg2[0];

  for (int tile = blockIdx.x * wpb + wave; tile < tiles;
       tile += gridDim.x * wpb) {
    const int e0 = tile << 4;
    int erow = e0 + m;
    if (erow >= n_edges) erow = n_edges - 1;
    const float* __restrict__ arow = edge_attr + (size_t)erow * EDGE_DIM;

    v2f a[16];
#pragma unroll
    for (int kk = 0; kk < 16; ++kk)
      a[kk] = *(const v2f*)(arow + kk * 4 + koff);

    float srow[8];
#pragma unroll
    for (int v = 0; v < 8; ++v) srow[v] = 0.0f;

#pragma unroll 1
    for (int nt = 0; nt < D_MODEL / 16; ++nt) {
      const int n = nt * 16 + m;
      v8f c = {0.f, 0.f, 0.f, 0.f, 0.f, 0.f, 0.f, 0.f};
#pragma unroll
      for (int kk = 0; kk < 16; ++kk) {
        v2f b = *(const v2f*)&Wt[n * LDS_STR_G + kk * 4 + koff];
        c = wmma_f32_16x16x4(a[kk], b, c);
      }
      const float bias = bg1[n];
      const float w2 = Wg2[n];
#pragma unroll
      for (int v = 0; v < 8; ++v) srow[v] += gelu_exact(c[v] + bias) * w2;
    }

    // Reduce over the 16 columns held by each lane half.
#pragma unroll
    for (int off = 8; off > 0; off >>= 1) {
#pragma unroll
      for (int v = 0; v < 8; ++v) srow[v] += __shfl_xor(srow[v], off, 32);
    }
    if (m == 0) {
#pragma unroll
      for (int v = 0; v < 8; ++v) {
        const int e = e0 + half * 8 + v;  // lane0 -> M 0..7, lane16 -> 8..15
        if (e < n_edges) gate[e] = sigmoidf(srow[v] + b2);
      }
    }
  }
}

// ---------------------------------------------------------------------------
// msg = (x_src[s] @ W_src) * gate ; scatter-add into acc[d]; count deg[d].
__global__ void k_msg_scatter(const float* __restrict__ x_src,
                              const int* __restrict__ s_idx,
                              const int* __restrict__ d_idx,
                              const float* __restrict__ W_src,
                              const float* __restrict__ gate,
                              float* __restrict__ acc,
                              float* __restrict__ deg, int n_edges) {
  __shared__ float Wt[D_MODEL * LDS_STR_W];  // Wt[n*130 + k] = W_src[k][n]
  const int tid = threadIdx.x;
  for (int i = tid; i < D_MODEL * D_MODEL; i += blockDim.x) {
    const int k = i >> 7, n = i & 127;
    Wt[n * LDS_STR_W + k] = W_src[i];
  }
  __syncthreads();

  const int lane = tid & 31;
  const int wave = tid >> 5;
  const int wpb = blockDim.x >> 5;
  const int tiles = (n_edges + 15) >> 4;
  const int m = lane & 15;
  const int half = lane >> 4;
  const int koff = half ? 2 : 0;

  for (int tile = blockIdx.x * wpb + wave; tile < tiles;
       tile += gridDim.x * wpb) {
    const int e0 = tile << 4;
    int erow = e0 + m;
    if (erow >= n_edges) erow = n_edges - 1;
    const int src = s_idx[erow];
    const float* __restrict__ arow = x_src + (size_t)src * D_MODEL;

    v2f a[32];  // full 16x128 A tile, read exactly once (L2-resident gather)
#pragma unroll
    for (int kk = 0; kk < 32; ++kk)
      a[kk] = *(const v2f*)(arow + kk * 4 + koff);

    float g[8];
    int dst[8];
    bool ok[8];
#pragma unroll
    for (int v = 0; v < 8; ++v) {
      const int e = e0 + half * 8 + v;  // C-tile row M=v(+8) -> this edge
      const int ec = (e < n_edges) ? e : (n_edges - 1);
      g[v] = gate[ec];
      dst[v] = d_idx[ec];
      ok[v] = (e < n_edges);
    }

#pragma unroll 1
    for (int nt = 0; nt < D_MODEL / 16; ++nt) {
      const int n = nt * 16 + m;
      v8f c = {0.f, 0.f, 0.f, 0.f, 0.f, 0.f, 0.f, 0.f};
#pragma unroll
      for (int kk = 0; kk < 32; ++kk) {
        v2f b = *(const v2f*)&Wt[n * LDS_STR_W + kk * 4 + koff];
        c = wmma_f32_16x16x4(a[kk], b, c);
      }
#pragma unroll
      for (int v = 0; v < 8; ++v) {
        if (ok[v])
          atomicAdd(&acc[(size_t)dst[v] * D_MODEL + n], c[v] * g[v]);
      }
    }

    if (half == 0 && (e0 + m) < n_edges)
      atomicAdd(&deg[d_idx[e0 + m]], 1.0f);
  }
}

// ---------------------------------------------------------------------------
// out[t] = x_dst[t] @ W_dst (bias added in finalize).
__global__ void k_dst_gemm(const float* __restrict__ x_dst,
                           const float* __restrict__ W_dst,
                           float* __restrict__ out, int n_nodes) {
  __shared__ float Wt[D_MODEL * LDS_STR_W];
  const int tid = threadIdx.x;
  for (int i = tid; i < D_MODEL * D_MODEL; i += blockDim.x) {
    const int k = i >> 7, n = i & 127;
    Wt[n * LDS_STR_W + k] = W_dst[i];
  }
  __syncthreads();

  const int lane = tid & 31;
  const int wave = tid >> 5;
  const int wpb = blockDim.x >> 5;
  const int tiles = (n_nodes + 15) >> 4;
  const int m = lane & 15;
  const int half = lane >> 4;
  const int koff = half ? 2 : 0;

  for (int tile = blockIdx.x * wpb + wave; tile < tiles;
       tile += gridDim.x * wpb) {
    const int t0 = tile << 4;
    int nrow = t0 + m;
    if (nrow >= n_nodes) nrow = n_nodes - 1;
    const float* __restrict__ arow = x_dst + (size_t)nrow * D_MODEL;

    v2f a[32];
#pragma unroll
    for (int kk = 0; kk < 32; ++kk)
      a[kk] = *(const v2f*)(arow + kk * 4 + koff);

#pragma unroll 1
    for (int nt = 0; nt < D_MODEL / 16; ++nt) {
      const int n = nt * 16 + m;
      v8f c = {0.f, 0.f, 0.f, 0.f, 0.f, 0.f, 0.f, 0.f};
#pragma unroll
      for (int kk = 0; kk < 32; ++kk) {
        v2f b = *(const v2f*)&Wt[n * LDS_STR_W + kk * 4 + koff];
        c = wmma_f32_16x16x4(a[kk], b, c);
      }
#pragma unroll
      for (int v = 0; v < 8; ++v) {
        const int row = t0 + half * 8 + v;
        if (row < n_nodes) out[(size_t)row * D_MODEL + n] = c[v];
      }
    }
  }
}

// ---------------------------------------------------------------------------
// out = gelu(LN(acc/max(deg,1) + out + b_dst)) ; one wave per node row.
__global__ void k_finalize(const float* __restrict__ acc,
                           const float* __restrict__ deg,
                           const float* __restrict__ b_dst,
                           const float* __restrict__ gamma,
                           const float* __restrict__ beta,
                           float* __restrict__ out, int n_nodes) {
  const int lane = threadIdx.x & 31;
  const int wave = threadIdx.x >> 5;
  const int row = blockIdx.x * (blockDim.x >> 5) + wave;
  if (row >= n_nodes) return;

  const int c0 = lane * 4;
  const float4 a = *(const float4*)(acc + (size_t)row * D_MODEL + c0);
  const float4 w = *(const float4*)(out + (size_t)row * D_MODEL + c0);
  const float4 b = *(const float4*)(b_dst + c0);
  const float invd = 1.0f / fmaxf(deg[row], 1.0f);

  float x0 = a.x * invd + w.x + b.x;
  float x1 = a.y * invd + w.y + b.y;
  float x2 = a.z * invd + w.z + b.z;
  float x3 = a.w * invd + w.w + b.w;

  float s = x0 + x1 + x2 + x3;
  float sq = x0 * x0 + x1 * x1 + x2 * x2 + x3 * x3;
#pragma unroll
  for (int off = 16; off > 0; off >>= 1) {
    s += __shfl_xor(s, off, 32);
    sq += __shfl_xor(sq, off, 32);
  }
  const float mean = s * (1.0f / (float)D_MODEL);
  const float var = sq * (1.0f / (float)D_MODEL) - mean * mean;
  const float rstd = rsqrtf(var + LN_EPS);

  const float4 gm = *(const float4*)(gamma + c0);
  const float4 bt = *(const float4*)(beta + c0);
  float4 o;
  o.x = gelu_exact((x0 - mean) * rstd * gm.x + bt.x);
  o.y = gelu_exact((x1 - mean) * rstd * gm.y + bt.y);
  o.z = gelu_exact((x2 - mean) * rstd * gm.z + bt.z);
  o.w = gelu_exact((x3 - mean) * rstd * gm.w + bt.w);
  *(float4*)(out + (size_t)row * D_MODEL + c0) = o;
}

// ---------------------------------------------------------------------------
extern "C" void kernel_launch(void* const* d_in, const int* in_sizes, int n_in,
                              void* d_out, int out_size, void* d_ws,
                              size_t ws_size, hipStream_t stream) {
  const float* x_src     = (const float*)d_in[0];
  const float* x_dst     = (const float*)d_in[1];
  const float* edge_attr = (const float*)d_in[2];
  const int*   edge_idx  = (const int*)d_in[3];
  const float* W_src     = (const float*)d_in[4];
  const float* W_dst     = (const float*)d_in[5];
  const float* b_dst     = (const float*)d_in[6];
  const float* Wg1       = (const float*)d_in[7];
  const float* bg1       = (const float*)d_in[8];
  const float* Wg2       = (const float*)d_in[9];
  const float* bg2       = (const float*)d_in[10];
  const float* gamma     = (const float*)d_in[11];
  const float* beta      = (const float*)d_in[12];

  const int n_nodes = in_sizes[0] / D_MODEL;
  const int n_edges = in_sizes[2] / EDGE_DIM;
  const int* s_idx = edge_idx;
  const int* d_idx = edge_idx + n_edges;

  // Workspace: acc[N,128] | deg[N] | gate[E]  (~56 MB at reference sizes)
  float* acc  = (float*)d_ws;
  float* deg  = acc + (size_t)n_nodes * D_MODEL;
  float* gate = deg + n_nodes;
  float* outf = (float*)d_out;

  // 1) zero accumulators (workspace is poisoned by the harness)
  {
    const long long nz = (long long)n_nodes * D_MODEL + n_nodes;
    const int bl = 256;
    k_zero<<<(int)((nz + bl - 1) / bl), bl, 0, stream>>>(acc, nz);
  }

  const int TPB = 256;  // 8 waves/block
  const int WPB = TPB / 32;
  const int etiles = (n_edges + 15) / 16;
  const int ntiles = (n_nodes + 15) / 16;

  // Persistent grids: enough waves to fill the chip, few enough that LDS
  // weight staging stays amortized (blocks * 64KB of W traffic total).
  int eblk = (etiles + WPB - 1) / WPB;
  if (eblk > 1024) eblk = 1024;
  int nblk = (ntiles + WPB - 1) / WPB;
  if (nblk > 512) nblk = 512;

  // 2) edge gate MLP (WMMA f32, Wg1 in LDS)
  k_edge_gate<<<eblk, TPB, 0, stream>>>(edge_attr, Wg1, bg1, Wg2, bg2, gate,
                                        n_edges);

  // 3) gather -> GEMM -> gate -> scatter-add (WMMA f32 + L2 atomics)
  k_msg_scatter<<<eblk, TPB, 0, stream>>>(x_src, s_idx, d_idx, W_src, gate,
                                          acc, deg, n_edges);

  // 4) x_dst @ W_dst into d_out (WMMA f32, W_dst in LDS)
  k_dst_gemm<<<nblk, TPB, 0, stream>>>(x_dst, W_dst, outf, n_nodes);

  // 5) mean-by-degree + residual + LN + gelu, in place on d_out
  k_finalize<<<(n_nodes + WPB - 1) / WPB, TPB, 0, stream>>>(
      acc, deg, b_dst, gamma, beta, outf, n_nodes);
}